// IntraAgg_28673201668651
// MI455X (gfx1250) — compile-verified
//
#include <hip/hip_runtime.h>
#include <hip/hip_bf16.h>

typedef float v2f __attribute__((ext_vector_type(2)));
typedef float v8f __attribute__((ext_vector_type(8)));

// ---------------------------------------------------------------------------
// Kernel 1: adaptive selection (stable smallest-32 of |center - neigh_score|)
// 256 threads = 4 rows x 64 candidates. Rank-counting reproduces the stable
// ordering of jax.lax.top_k(-diff): ascending diff, ties by original index.
// Writes samp_scores[b*32+rank] and selected node ids to workspace.
// ---------------------------------------------------------------------------
__global__ __launch_bounds__(256)
void IntraAgg_select_kernel(const float* __restrict__ center_scores,
                            const float* __restrict__ neigh_scores,
                            const int*   __restrict__ neighs,
                            float* __restrict__ samp_scores,   // [B,32]
                            int*   __restrict__ sel_ids,       // [B,32] (ws)
                            int B) {
    __shared__ float diffs[4][64];
    const int lrow = threadIdx.x >> 6;      // 0..3
    const int k    = threadIdx.x & 63;      // candidate
    const int row  = blockIdx.x * 4 + lrow;
    if (row >= B) return;

    const float c = center_scores[row];
    const float d = fabsf(c - neigh_scores[row * 64 + k]);
    diffs[lrow][k] = d;
    __syncthreads();

    const float* dr = diffs[lrow];
    int rank = 0;
#pragma unroll
    for (int j = 0; j < 64; ++j) {
        const float dj = dr[j];
        rank += (dj < d || (dj == d && j < k)) ? 1 : 0;
    }
    if (rank < 32) {
        samp_scores[row * 32 + rank] = d;
        sel_ids[row * 32 + rank]     = neighs[row * 64 + k];
    }
}

// ---------------------------------------------------------------------------
// Kernel 2: gather + mean + relu via V_WMMA_F32_16X16X4_F32 (full f32 precision).
// One wave per tile of 16 batch rows: D[16,64] = A[16,512] x E[512,64],
// A block-diagonal = 1/32. K-loop: 128 chunks of 4 samples; all 4 samples of a
// chunk belong to row (t>>3), so the A fragment is uniform {aval,aval}.
// B fragment: lanes 0-15 carry K=0/1 rows, lanes 16-31 K=2/3 rows, col=lane&15.
// Four N-tile accumulators so each gathered 256B feature row is consumed
// back-to-back (single pass over memory, element loaded exactly once).
// ---------------------------------------------------------------------------
__global__ __launch_bounds__(256)
void IntraAgg_wmma_agg_kernel(const float* __restrict__ features,
                              const int*   __restrict__ sel_ids,  // [B,32]
                              float* __restrict__ to_feats,       // [B,64]
                              int n_tiles) {
    __shared__ int ids[8][512];             // per-wave staging, 16 KB
    const int wv   = threadIdx.x >> 5;      // wave in block: 0..7
    const int lane = threadIdx.x & 31;
    const int tile = blockIdx.x * 8 + wv;
    if (tile >= n_tiles) return;

    // Stage this tile's 512 selected node ids into LDS (coalesced 128B loads).
    const int base = tile * 512;
#pragma unroll
    for (int i = 0; i < 16; ++i)
        ids[wv][i * 32 + lane] = sel_ids[base + i * 32 + lane];
    __syncthreads();

    const int hf  = lane >> 4;              // 0: K=0/1 half, 1: K=2/3 half
    const int col = lane & 15;              // N within 16-wide tile / A's M

    v8f acc0 = {}, acc1 = {}, acc2 = {}, acc3 = {};

    for (int t = 0; t < 128; ++t) {
        const int i0 = ids[wv][t * 4 + hf * 2 + 0];   // K = hf*2
        const int i1 = ids[wv][t * 4 + hf * 2 + 1];   // K = hf*2 + 1
        const float* r0 = features + (size_t)i0 * 64 + col;
        const float* r1 = features + (size_t)i1 * 64 + col;

        const float aval = (col == (t >> 3)) ? 0.03125f : 0.0f; // 1/32 exact
        v2f a = {aval, aval};

        v2f b0 = { r0[0],  r1[0]  };
        v2f b1 = { r0[16], r1[16] };
        v2f b2 = { r0[32], r1[32] };
        v2f b3 = { r0[48], r1[48] };

        acc0 = __builtin_amdgcn_wmma_f32_16x16x4_f32(false, a, false, b0,
                                                     (short)0, acc0, false, false);
        acc1 = __builtin_amdgcn_wmma_f32_16x16x4_f32(false, a, false, b1,
                                                     (short)0, acc1, false, false);
        acc2 = __builtin_amdgcn_wmma_f32_16x16x4_f32(false, a, false, b2,
                                                     (short)0, acc2, false, false);
        acc3 = __builtin_amdgcn_wmma_f32_16x16x4_f32(false, a, false, b3,
                                                     (short)0, acc3, false, false);
    }

    // D layout: VGPR r -> M = hf*8 + r, N = ntile*16 + col.  relu + store.
    const int mbase = tile * 16 + hf * 8;
#pragma unroll
    for (int r = 0; r < 8; ++r) {
        float* orow = to_feats + (size_t)(mbase + r) * 64 + col;
        orow[0]  = fmaxf(acc0[r], 0.0f);
        orow[16] = fmaxf(acc1[r], 0.0f);
        orow[32] = fmaxf(acc2[r], 0.0f);
        orow[48] = fmaxf(acc3[r], 0.0f);
    }
}

// ---------------------------------------------------------------------------
// Launch. d_out = [to_feats (B*64) | samp_scores (B*32)] floats.
// ---------------------------------------------------------------------------
extern "C" void kernel_launch(void* const* d_in, const int* in_sizes, int n_in,
                              void* d_out, int out_size, void* d_ws, size_t ws_size,
                              hipStream_t stream) {
    const float* features      = (const float*)d_in[0];
    const float* center_scores = (const float*)d_in[1];
    const float* neigh_scores  = (const float*)d_in[2];
    const int*   neighs        = (const int*)d_in[3];
    const int B = in_sizes[1];              // center_scores is [B,1]

    float* to_feats    = (float*)d_out;
    float* samp_scores = (float*)d_out + (size_t)B * 64;
    int*   sel_ids     = (int*)d_ws;        // B*32 ints of scratch

    IntraAgg_select_kernel<<<(B + 3) / 4, 256, 0, stream>>>(
        center_scores, neigh_scores, neighs, samp_scores, sel_ids, B);

    const int n_tiles = B / 16;
    IntraAgg_wmma_agg_kernel<<<(n_tiles + 7) / 8, 256, 0, stream>>>(
        features, sel_ids, to_feats, n_tiles);
}